// StreamingEncoder_35527969472868
// MI455X (gfx1250) — compile-verified
//
#include <hip/hip_runtime.h>
#include <math.h>

typedef __attribute__((ext_vector_type(2))) float v2f;
typedef __attribute__((ext_vector_type(8))) float v8f;

#define NTHREADS 512
#define NWAVES   16

// ---- LDS layout (floats). Activations stored with 1-column zero halo on each
// side: buffer shape [C][W+2], interior at col+1. Whole smem zeroed once so
// halos stay zero. ----
#define AW0     0        // 64 x 8   zero-padded W_erb0 (K 6 -> 8)
#define CTX_ERB 512      // 2 x 34
#define CTX_DF  580      // 4 x 98   (row = ch*2 + kt)
#define E0      972      // 64 x 34
#define E1      3148     // 64 x 18
#define E2      4300     // 64 x 10
#define E3      4940     // 64 x 6
#define C0      5324     // 64 x 98
#define C1      11596    // 64 x 50
#define C1F     972      // 3072 (reuses dead E0/E1)
#define CEMB    5324     // 256  (reuses dead C0)
#define X2      5580     // 256
#define GI      5836     // 768
#define GH      6604     // 768
#define HN      7372     // 256
#define SMEM_F  14848

// ---- output offsets (flat concat, reference return order) ----
#define OUT_E0   0
#define OUT_E1   2048
#define OUT_E2   3072
#define OUT_E3   3584
#define OUT_EMB  3840
#define OUT_C0   4096
#define OUT_LSNR 10240
#define OUT_BERB 10241
#define OUT_BDF  10273
#define OUT_H    10465

__device__ __forceinline__ float wave_reduce_add(float v) {
  #pragma unroll
  for (int off = 16; off > 0; off >>= 1) v += __shfl_xor(v, off, 32);
  return v;
}

__device__ __forceinline__ float sigm(float x) { return 1.0f / (1.0f + __expf(-x)); }

// GEMM via WMMA f32 16x16x4: D = relu(W (Cout x K) * im2col(src) + bias).
// src is halo-padded LDS [rows][Win+2]; im2col column n, row k maps to
// (r = k/3, kw = k%3) -> src[r*(Win+2) + STRIDE*n + kw]  (fully branch-free).
// dstLds is halo-padded [Cout][Nout+2]; dstG (if STORE_G) compact [Cout][Nout].
template <int Cout, int K, int Win, int Nout, int STRIDE, bool STORE_G>
__device__ void conv_wmma(const float* __restrict__ Wg, const float* __restrict__ bias,
                          const float* src, float* dstLds, float* dstG,
                          int wave, int lane) {
  constexpr int numM = Cout >> 4;
  constexpr int numN = (Nout + 15) >> 4;
  constexpr int ntiles = numM * numN;
  constexpr int WP = Win + 2;
  constexpr int NP = Nout + 2;
  const int ln = lane & 15;
  const int half = lane >> 4;
  for (int t = wave; t < ntiles; t += NWAVES) {
    const int m0 = (t / numN) << 4;
    const int n0 = (t % numN) << 4;
    v8f c = {0.f, 0.f, 0.f, 0.f, 0.f, 0.f, 0.f, 0.f};
    const int arow = m0 + ln;
    const int ncol = n0 + ln;
    #pragma unroll 8
    for (int k0 = 0; k0 < K; k0 += 4) {
      const int kb = k0 + 2 * half;
      const float2 av = *(const float2*)(Wg + arow * K + kb);  // 8B aligned: K, kb even
      v2f a, b;
      a.x = av.x;
      a.y = av.y;
      #pragma unroll
      for (int j = 0; j < 2; ++j) {
        const int k = kb + j;
        const int r = k / 3;
        const int kw = k - 3 * r;
        b[j] = src[r * WP + STRIDE * ncol + kw];
      }
      c = __builtin_amdgcn_wmma_f32_16x16x4_f32(false, a, false, b, (short)0, c, false, false);
    }
    bool ok = true;
    if constexpr ((Nout & 15) != 0) ok = (ncol < Nout);
    if (ok) {
      const int mb = m0 + 8 * half;                 // multiple of 8 -> 32B aligned
      const float4 ba = *(const float4*)(bias + mb);
      const float4 bb = *(const float4*)(bias + mb + 4);
      const float bv[8] = {ba.x, ba.y, ba.z, ba.w, bb.x, bb.y, bb.z, bb.w};
      #pragma unroll
      for (int r = 0; r < 8; ++r) {
        float v = c[r] + bv[r];
        v = v > 0.f ? v : 0.f;
        dstLds[(mb + r) * NP + 1 + ncol] = v;
        if constexpr (STORE_G) dstG[(mb + r) * Nout + ncol] = v;
      }
    }
  }
}

// out[m] = (relu?) (W[m,:] . x + b[m]); K % 128 == 0; float4 loads.
__device__ void gemv(const float* __restrict__ W, const float* __restrict__ b,
                     const float* x, float* out, int M, int K,
                     int wave, int lane, bool do_relu) {
  for (int m = wave; m < M; m += NWAVES) {
    float s = 0.f;
    for (int k = lane * 4; k < K; k += 128) {
      const float4 wv = *(const float4*)(W + m * K + k);
      const float4 xv = *(const float4*)(x + k);
      s += wv.x * xv.x + wv.y * xv.y + wv.z * xv.z + wv.w * xv.w;
    }
    s = wave_reduce_add(s);
    if (lane == 0) {
      float v = s + b[m];
      if (do_relu) v = v > 0.f ? v : 0.f;
      out[m] = v;
    }
  }
}

__global__ __launch_bounds__(NTHREADS, 1)
void StreamingEncoder_35527969472868_kernel(
    const float* __restrict__ feat_erb, const float* __restrict__ feat_spec,
    const float* __restrict__ buf_erb0, const float* __restrict__ buf_df0,
    const float* __restrict__ h_enc,
    const float* __restrict__ W_erb0, const float* __restrict__ b_erb0,
    const float* __restrict__ W_erb1, const float* __restrict__ b_erb1,
    const float* __restrict__ W_erb2, const float* __restrict__ b_erb2,
    const float* __restrict__ W_erb3, const float* __restrict__ b_erb3,
    const float* __restrict__ W_df0,  const float* __restrict__ b_df0,
    const float* __restrict__ W_df1,  const float* __restrict__ b_df1,
    const float* __restrict__ W_dffc, const float* __restrict__ b_dffc,
    const float* __restrict__ W_ih,   const float* __restrict__ b_ih,
    const float* __restrict__ W_hh,   const float* __restrict__ b_hh,
    const float* __restrict__ W_lsnr, const float* __restrict__ b_lsnr,
    float* __restrict__ out) {
  __shared__ float smem[SMEM_F];
  const int tid  = threadIdx.x;
  const int wave = tid >> 5;
  const int lane = tid & 31;

  // Prefetch the big GEMV weights (~4.7MB) into L2 so HBM fetch overlaps convs.
  for (int i = tid * 64; i < 256 * 3072; i += NTHREADS * 64)
    __builtin_prefetch(W_dffc + i, 0, 1);
  for (int i = tid * 64; i < 768 * 256; i += NTHREADS * 64) {
    __builtin_prefetch(W_ih + i, 0, 1);
    __builtin_prefetch(W_hh + i, 0, 1);
  }

  // Zero all of LDS once: establishes the zero halos and K-padding.
  for (int i = tid; i < SMEM_F; i += NTHREADS) smem[i] = 0.f;
  __syncthreads();

  // Stage 0: contexts into LDS (interior of halo layout), padded W_erb0,
  // and pass-through buffer outputs.
  for (int i = tid; i < 384; i += NTHREADS) {          // W_erb0: 64 x 6 -> 64 x 8
    const int m = i / 6, k = i - m * 6;
    smem[AW0 + m * 8 + k] = W_erb0[i];
  }
  for (int i = tid; i < 32; i += NTHREADS) {
    const float v = feat_erb[i];
    smem[CTX_ERB + 34 + 1 + i] = v;                    // kt=1
    smem[CTX_ERB + 1 + i] = buf_erb0[i];               // kt=0
    out[OUT_BERB + i] = v;                             // buf_erb0_new = feat_erb
  }
  for (int i = tid; i < 192; i += NTHREADS) {
    const int ch = i / 96, w = i - ch * 96;
    const float v = feat_spec[i];
    smem[CTX_DF + (ch * 2 + 1) * 98 + 1 + w] = v;          // kt=1
    smem[CTX_DF + (ch * 2 + 0) * 98 + 1 + w] = buf_df0[i]; // kt=0
    out[OUT_BDF + i] = v;                                  // buf_df0_new = feat_spec
  }
  __syncthreads();

  // ERB path
  conv_wmma<64,   8, 32, 32, 1, true >(smem + AW0, b_erb0, smem + CTX_ERB, smem + E0, out + OUT_E0, wave, lane);
  __syncthreads();
  conv_wmma<64, 192, 32, 16, 2, true >(W_erb1, b_erb1, smem + E0, smem + E1, out + OUT_E1, wave, lane);
  __syncthreads();
  conv_wmma<64, 192, 16,  8, 2, true >(W_erb2, b_erb2, smem + E1, smem + E2, out + OUT_E2, wave, lane);
  __syncthreads();
  conv_wmma<64, 192,  8,  4, 2, true >(W_erb3, b_erb3, smem + E2, smem + E3, out + OUT_E3, wave, lane);
  __syncthreads();

  // DF path
  conv_wmma<64,  12, 96, 96, 1, true >(W_df0, b_df0, smem + CTX_DF, smem + C0, out + OUT_C0, wave, lane);
  __syncthreads();
  conv_wmma<64, 192, 96, 48, 2, false>(W_df1, b_df1, smem + C0, smem + C1, nullptr, wave, lane);
  __syncthreads();

  // c1f = transpose(c1) -> [pos*64 + ch]
  for (int i = tid; i < 3072; i += NTHREADS)
    smem[C1F + i] = smem[C1 + (i & 63) * 50 + 1 + (i >> 6)];
  __syncthreads();

  // cemb = relu(W_dffc @ c1f + b)   (overwrites dead C0 region)
  gemv(W_dffc, b_dffc, smem + C1F, smem + CEMB, 256, 3072, wave, lane, true);
  __syncthreads();

  // x2 = flatten(e3, NHWC) + cemb
  for (int i = tid; i < 256; i += NTHREADS) {
    const int p = i >> 6, ch = i & 63;
    smem[X2 + i] = smem[E3 + ch * 6 + 1 + p] + smem[CEMB + i];
  }
  __syncthreads();
  // stash h2 into CEMB region (cemb no longer needed)
  for (int i = tid; i < 256; i += NTHREADS) smem[CEMB + i] = h_enc[i];
  __syncthreads();

  // GRU gates
  gemv(W_ih, b_ih, smem + X2,   smem + GI, 768, 256, wave, lane, false);
  gemv(W_hh, b_hh, smem + CEMB, smem + GH, 768, 256, wave, lane, false);
  __syncthreads();

  for (int i = tid; i < 256; i += NTHREADS) {
    const float r = sigm(smem[GI + i]       + smem[GH + i]);
    const float z = sigm(smem[GI + 256 + i] + smem[GH + 256 + i]);
    const float n = tanhf(smem[GI + 512 + i] + r * smem[GH + 512 + i]);
    const float h2v = smem[CEMB + i];
    const float hn = (1.f - z) * n + z * h2v;
    smem[HN + i] = hn;
    out[OUT_EMB + i] = hn;   // emb
    out[OUT_H + i]   = hn;   // h_enc_new
  }
  __syncthreads();

  // lsnr head
  if (wave == 0) {
    float s = 0.f;
    for (int k = lane * 4; k < 256; k += 128) {
      const float4 wv = *(const float4*)(W_lsnr + k);
      const float4 xv = *(const float4*)(smem + HN + k);
      s += wv.x * xv.x + wv.y * xv.y + wv.z * xv.z + wv.w * xv.w;
    }
    s = wave_reduce_add(s);
    if (lane == 0)
      out[OUT_LSNR] = sigm(s + b_lsnr[0]) * 50.0f + (-15.0f);
  }
}

extern "C" void kernel_launch(void* const* d_in, const int* in_sizes, int n_in,
                              void* d_out, int out_size, void* d_ws, size_t ws_size,
                              hipStream_t stream) {
  (void)in_sizes; (void)n_in; (void)out_size; (void)d_ws; (void)ws_size;
  StreamingEncoder_35527969472868_kernel<<<dim3(1), dim3(NTHREADS), 0, stream>>>(
      (const float*)d_in[0],  (const float*)d_in[1],  (const float*)d_in[2],
      (const float*)d_in[3],  (const float*)d_in[4],
      (const float*)d_in[5],  (const float*)d_in[6],
      (const float*)d_in[7],  (const float*)d_in[8],
      (const float*)d_in[9],  (const float*)d_in[10],
      (const float*)d_in[11], (const float*)d_in[12],
      (const float*)d_in[13], (const float*)d_in[14],
      (const float*)d_in[15], (const float*)d_in[16],
      (const float*)d_in[17], (const float*)d_in[18],
      (const float*)d_in[19], (const float*)d_in[20],
      (const float*)d_in[21], (const float*)d_in[22],
      (const float*)d_in[23], (const float*)d_in[24],
      (float*)d_out);
}